// DeuantizationLayer_45715631898660
// MI455X (gfx1250) — compile-verified
//
#include <hip/hip_runtime.h>

// Bit2Num dequantization for MI455X (gfx1250, wave32).
//
// Memory-bound (320 MB traffic, ~13.7 us floor at 23.3 TB/s). The K=4 weighted
// sum maps exactly onto V_WMMA_F32_16X16X4_F32:
//   A (16x4)  = weights/16 broadcast down rows (lane-constant, 2 VGPRs)
//   B (4x16)  = bits of 16 consecutive groups (one coalesced b64 load per lane)
//   C (16x16) = 0.03125 broadcast  =>  D = (num + 0.5)/16  (final output)
// All rows of D are equal, so d[0] holds the 16 results in lanes 0-15 AND
// lanes 16-31; pairing two tiles per store gives a flat coalesced b32 store.
// Non-temporal loads/stores keep the 320 MB one-shot stream out of L2.

typedef __attribute__((ext_vector_type(2))) float v2f;
typedef __attribute__((ext_vector_type(8))) float v8f;

#if defined(__HIP_DEVICE_COMPILE__) && __has_builtin(__builtin_amdgcn_wmma_f32_16x16x4_f32)
#define B2N_USE_WMMA 1
#else
#define B2N_USE_WMMA 0
#endif

__global__ __launch_bounds__(256) void bit2num_wmma_kernel(
    const float* __restrict__ x, float* __restrict__ out, int n_groups) {
#if B2N_USE_WMMA
  const int lane = threadIdx.x & 31;
  const int half = lane >> 4;                       // 0: lanes 0-15, 1: lanes 16-31
  const int wave = blockIdx.x * 8 + (threadIdx.x >> 5);
  const long t0  = (long)wave * 4;                  // first 16-group tile of wave

  const long totalFloats = (long)n_groups * 4;
  const long maxLd = totalFloats - 2;               // last valid (even) v2f start

  // A-matrix: A[m][k] = w[k]/16, row-constant. 32-bit 16x4 layout:
  // lanes 0-15 hold K=0 (v0), K=1 (v1); lanes 16-31 hold K=2 (v0), K=3 (v1).
  v2f a;
  a.x = half ? 0.125f  : 0.5f;    // 2/16 : 8/16
  a.y = half ? 0.0625f : 0.25f;   // 1/16 : 4/16

  // C-matrix: 0.5/16 broadcast.
  v8f c = {0.03125f, 0.03125f, 0.03125f, 0.03125f,
           0.03125f, 0.03125f, 0.03125f, 0.03125f};

  // B-matrix load: lane l takes the contiguous float pair at
  // tile*64 + 4*(l&15) + 2*(l>>4)  -> one fully coalesced b64 per lane.
  const long off = (long)(4 * (lane & 15) + 2 * half);
  long i0 = t0 * 64 + off;
  long i1 = i0 + 64;
  long i2 = i0 + 128;
  long i3 = i0 + 192;
  i0 = (i0 > maxLd) ? maxLd : i0;   // clamp (tail safety; stays 8B-aligned)
  i1 = (i1 > maxLd) ? maxLd : i1;
  i2 = (i2 > maxLd) ? maxLd : i2;
  i3 = (i3 > maxLd) ? maxLd : i3;

  v2f b0 = __builtin_nontemporal_load((const v2f*)(x + i0));
  v2f b1 = __builtin_nontemporal_load((const v2f*)(x + i1));
  v2f b2 = __builtin_nontemporal_load((const v2f*)(x + i2));
  v2f b3 = __builtin_nontemporal_load((const v2f*)(x + i3));

  // EXEC is all-1s here: no divergent branches before/through the WMMAs.
  v8f d0 = __builtin_amdgcn_wmma_f32_16x16x4_f32(false, a, false, b0, (short)0, c, false, false);
  v8f d1 = __builtin_amdgcn_wmma_f32_16x16x4_f32(false, a, false, b1, (short)0, c, false, false);
  v8f d2 = __builtin_amdgcn_wmma_f32_16x16x4_f32(false, a, false, b2, (short)0, c, false, false);
  v8f d3 = __builtin_amdgcn_wmma_f32_16x16x4_f32(false, a, false, b3, (short)0, c, false, false);

  // d[0]: lanes 0-15 = D[0][n] (groups n of tile), lanes 16-31 = D[8][n]
  // (identical, since all rows of D are equal). Pair tiles {even,odd} so the
  // store address is flat: (t0+2s)*16 + lane.
  const float r0 = half ? d1[0] : d0[0];
  const float r1 = half ? d3[0] : d2[0];

  long o0 = t0 * 16 + lane;
  long o1 = o0 + 32;
  if (o0 < n_groups) __builtin_nontemporal_store(r0, out + o0);
  if (o1 < n_groups) __builtin_nontemporal_store(r1, out + o1);
#else
  // Fallback (builtin unavailable): identical group mapping, plain FMA chain.
  const int lane = threadIdx.x & 31;
  const int wave = blockIdx.x * 8 + (threadIdx.x >> 5);
  long o0 = (long)wave * 64 + lane;
  long o1 = o0 + 32;
  for (int s = 0; s < 2; ++s) {
    long o = s ? o1 : o0;
    if (o < n_groups) {
      const float* g = x + o * 4;
      float v0 = __builtin_nontemporal_load(g + 0);
      float v1 = __builtin_nontemporal_load(g + 1);
      float v2 = __builtin_nontemporal_load(g + 2);
      float v3 = __builtin_nontemporal_load(g + 3);
      float r = fmaf(0.5f, v0,
                fmaf(0.25f, v1,
                fmaf(0.125f, v2,
                fmaf(0.0625f, v3, 0.03125f))));
      __builtin_nontemporal_store(r, out + o);
    }
  }
#endif
}

extern "C" void kernel_launch(void* const* d_in, const int* in_sizes, int n_in,
                              void* d_out, int out_size, void* d_ws, size_t ws_size,
                              hipStream_t stream) {
  (void)in_sizes; (void)n_in; (void)d_ws; (void)ws_size;
  // d_in[0]: x (float32, n_groups*4 elements). d_in[1]: B (==4, fixed by the
  // reference setup; the K=4 WMMA shape hardwires it).
  const float* x = (const float*)d_in[0];
  float* out = (float*)d_out;
  const int n_groups = out_size;                    // 1024 * 16384 = 16,777,216

  // 8 waves/block * 4 tiles/wave * 16 groups/tile = 512 groups per block.
  const int groups_per_block = 512;
  const int grid = (n_groups + groups_per_block - 1) / groups_per_block;
  bit2num_wmma_kernel<<<grid, 256, 0, stream>>>(x, out, n_groups);
}